// VolumeSDFRenderer_71227737637500
// MI455X (gfx1250) — compile-verified
//
#include <hip/hip_runtime.h>

typedef __attribute__((ext_vector_type(2))) float v2f;
typedef __attribute__((ext_vector_type(4))) float v4f;
typedef __attribute__((ext_vector_type(8))) float v8f;

#define ALPHA_C   10.0f
#define INV_BETA  20.0f      // 1/beta = 1/0.05
#define EPS_C     1e-10f
#define FAR_DELTA_C 1e10f

#define R_TOT 65536
#define NP    128
#define RAYS_PER_WAVE   16
#define WAVES_PER_BLOCK 2
#define RAYS_PER_BLOCK  (RAYS_PER_WAVE * WAVES_PER_BLOCK)
#define THREADS         (32 * WAVES_PER_BLOCK)

__global__ __launch_bounds__(THREADS)
void volsdf_render_kernel(const float* __restrict__ distance,
                          const float* __restrict__ color,
                          const float* __restrict__ lengths,
                          float* __restrict__ out)
{
    // Per-wave tiles: L = log(exp(-density*delta)+eps), W = weights
    __shared__ float sL[WAVES_PER_BLOCK][RAYS_PER_WAVE * NP];
    __shared__ float sW[WAVES_PER_BLOCK][RAYS_PER_WAVE * NP];

    const int tid  = threadIdx.x;
    const int wave = tid >> 5;
    const int lane = tid & 31;
    const int n    = lane & 15;   // column index within 16-wide tiles
    const int half = lane >> 4;   // lane group (K/M split per WMMA layouts)

    const int rblock = blockIdx.x * RAYS_PER_BLOCK;

    // ---------------- phase 0: zero geometry output slice ----------------
    // geometry = zeros_like(color): [R, 128, 3] right after out_color [R,3].
    // Non-temporal stores: keep the 96MB zero stream from evicting the
    // 160MB input working set (fits the 192MB L2 across graph replays).
    {
        v4f* g4 = (v4f*)(out + (size_t)R_TOT * 3 + (size_t)rblock * NP * 3);
        const v4f z = {0.f, 0.f, 0.f, 0.f};
        #pragma unroll
        for (int i = 0; i < (RAYS_PER_BLOCK * NP * 3) / (4 * THREADS); ++i)
            __builtin_nontemporal_store(z, g4 + tid + i * THREADS);
    }

    float* sLw = sL[wave];
    float* sWw = sW[wave];
    const int rwave = rblock + wave * RAYS_PER_WAVE;

    // ------- phase 1: compute L[ray][p] = log(exp(-density*delta)+eps) -------
    for (int r = 0; r < RAYS_PER_WAVE; ++r) {
        const size_t row = (size_t)(rwave + r) * NP;
        float4 d4 = *(const float4*)(distance + row + lane * 4);
        float4 t4 = *(const float4*)(lengths  + row + lane * 4);
        float nxt = __shfl_down(t4.x, 1, 32);   // lengths[4*lane+4]
        float dl0 = t4.y - t4.x;
        float dl1 = t4.z - t4.y;
        float dl2 = t4.w - t4.z;
        float dl3 = (lane == 31) ? FAR_DELTA_C : (nxt - t4.w);

        float den0 = ALPHA_C / (1.f + __expf(INV_BETA * d4.x));
        float den1 = ALPHA_C / (1.f + __expf(INV_BETA * d4.y));
        float den2 = ALPHA_C / (1.f + __expf(INV_BETA * d4.z));
        float den3 = ALPHA_C / (1.f + __expf(INV_BETA * d4.w));

        float4 Lv;
        Lv.x = __logf(__expf(-den0 * dl0) + EPS_C);
        Lv.y = __logf(__expf(-den1 * dl1) + EPS_C);
        Lv.z = __logf(__expf(-den2 * dl2) + EPS_C);
        Lv.w = __logf(__expf(-den3 * dl3) + EPS_C);
        *(float4*)(sLw + r * NP + lane * 4) = Lv;
    }
    __syncthreads();

    // ------- phase 2: exclusive cumsum via causal WMMA matmul -------
    // S[m, p] = sum_{k < p} L[m, k]  for 16 rays x 16 columns per tile.
    // A (16x4 f32): lane m=lane&15, half selects K+{0,1} vs K+{2,3}
    // B (4x16 f32): causal mask (k < p), generated in registers
    for (int c = 0; c < NP / 16; ++c) {
        v8f acc = {0.f, 0.f, 0.f, 0.f, 0.f, 0.f, 0.f, 0.f};
        const int p = c * 16 + n;          // global column this lane owns in D
        const int kend = 16 * c + 16;      // uniform per wave -> EXEC all ones
        for (int k0 = 0; k0 < kend; k0 += 4) {
            const int kk = k0 + 2 * half;
            v2f a = *(const v2f*)(sLw + n * NP + kk);  // A row = ray n
            v2f b;
            b.x = (kk     < p) ? 1.f : 0.f;
            b.y = (kk + 1 < p) ? 1.f : 0.f;
            acc = __builtin_amdgcn_wmma_f32_16x16x4_f32(
                false, a, false, b, (short)0, acc, false, false);
        }
        // D layout: lanes 0-15 -> M=g, lanes 16-31 -> M=g+8; N = lane&15
        #pragma unroll
        for (int g = 0; g < 8; ++g) {
            const int ray = g + 8 * half;
            float Lv = sLw[ray * NP + p];
            // trans = exp(S); alpha = 1 - exp(-a) = 1 + eps - exp(L)
            float w = __expf(acc[g]) * (1.f + EPS_C - __expf(Lv));
            sWw[ray * NP + p] = w;
        }
    }
    __syncthreads();

    // ------- phase 3: out_color[r, c] = sum_p w[r,p] * color[r,p,c] -------
    for (int r = 0; r < RAYS_PER_WAVE; ++r) {
        const int ray = rwave + r;
        float4 wv = *(const float4*)(sWw + r * NP + lane * 4);
        const float* cb = color + (size_t)ray * (NP * 3) + lane * 12;
        float4 f0 = *(const float4*)(cb + 0);
        float4 f1 = *(const float4*)(cb + 4);
        float4 f2 = *(const float4*)(cb + 8);
        // element (p=4l+j, ch) lives at f[3j+ch]
        float s0 = wv.x * f0.x + wv.y * f0.w + wv.z * f1.z + wv.w * f2.y;
        float s1 = wv.x * f0.y + wv.y * f1.x + wv.z * f1.w + wv.w * f2.z;
        float s2 = wv.x * f0.z + wv.y * f1.y + wv.z * f2.x + wv.w * f2.w;
        #pragma unroll
        for (int off = 16; off > 0; off >>= 1) {
            s0 += __shfl_xor(s0, off, 32);
            s1 += __shfl_xor(s1, off, 32);
            s2 += __shfl_xor(s2, off, 32);
        }
        if (lane == 0) {
            __builtin_nontemporal_store(s0, out + ray * 3 + 0);
            __builtin_nontemporal_store(s1, out + ray * 3 + 1);
            __builtin_nontemporal_store(s2, out + ray * 3 + 2);
        }
    }
}

extern "C" void kernel_launch(void* const* d_in, const int* in_sizes, int n_in,
                              void* d_out, int out_size, void* d_ws, size_t ws_size,
                              hipStream_t stream) {
    const float* distance = (const float*)d_in[0];  // [R, 128]
    const float* color    = (const float*)d_in[1];  // [R, 128, 3]
    const float* lengths  = (const float*)d_in[2];  // [R, 128]
    float* out = (float*)d_out;                     // [R,3] ++ [R,128,3] zeros

    dim3 grid(R_TOT / RAYS_PER_BLOCK);  // 2048 blocks
    dim3 block(THREADS);                // 64 threads (2 waves)
    volsdf_render_kernel<<<grid, block, 0, stream>>>(distance, color, lengths, out);
}